// FullDatasetCombinedLoss_12481174962336
// MI455X (gfx1250) — compile-verified
//
#include <hip/hip_runtime.h>
#include <hip/hip_bf16.h>
#include <math.h>

typedef float v2f __attribute__((ext_vector_type(2)));
typedef float v8f __attribute__((ext_vector_type(8)));

#define N_ROWS 4096
#define D_LAT  768
#define D_IN   1024
#define POS_MARGIN 2.0f
#define NEG_MARGIN 10.0f

#define BM 128
#define BN 128
#define BK 32
#define LSTR (BK + 2)   // pad so (row*34 + k) hits distinct banks for 16 consecutive rows

// d_ws layout:
//   doubles[0..4]: pos_sum, pos_cnt, neg_sum, neg_cnt, mse_sum
//   byte offset 64: float sq[N_ROWS]

__global__ void zero_accum_kernel(double* acc) {
    if (threadIdx.x < 8) acc[threadIdx.x] = 0.0;
}

__global__ __launch_bounds__(256) void row_sqnorm_kernel(
    const float* __restrict__ X, float* __restrict__ sq) {
    __shared__ float red[256];
    const int row = blockIdx.x;
    const float* xr = X + (size_t)row * D_LAT;
    float s = 0.0f;
    for (int c = threadIdx.x; c < D_LAT; c += 256) {
        float v = xr[c];
        s = fmaf(v, v, s);
    }
    red[threadIdx.x] = s;
    __syncthreads();
    for (int off = 128; off > 0; off >>= 1) {
        if (threadIdx.x < off) red[threadIdx.x] += red[threadIdx.x + off];
        __syncthreads();
    }
    if (threadIdx.x == 0) sq[row] = red[0];
}

#define WSTEP(CC, AA, BB) \
    CC = __builtin_amdgcn_wmma_f32_16x16x4_f32(false, AA, false, BB, (short)0, CC, false, false)

// 128x128 output tile per 256-thread block. 8 waves = 4 row-groups x 2 col-groups;
// each wave owns a 32x64 sub-tile = 8 f32 16x16 WMMA accumulators (2 A-frags x 4 B-frags
// per K-step of 4 -> 8 independent accumulation chains, 8 WMMAs per 6 two-float LDS loads).
// Only bi <= bj blocks run; epilogue keeps strictly-upper (i<j) elements, which is exactly
// half of the symmetric off-diagonal masked sums -> the masked means are unchanged.
__global__ __launch_bounds__(256) void gram_loss_kernel(
    const float* __restrict__ X, const float* __restrict__ sq,
    const int* __restrict__ labels, double* __restrict__ acc) {
    const int bj = blockIdx.x;
    const int bi = blockIdx.y;
    if (bi > bj) return;

    __shared__ float sA[BM * LSTR];
    __shared__ float sB[BN * LSTR];
    __shared__ float s_sqi[BM], s_sqj[BN];
    __shared__ int   s_li[BM],  s_lj[BN];
    __shared__ float red_x[256], red_y[256], red_z[256], red_w[256];

    const int t    = threadIdx.x;
    const int lane = t & 31;
    const int wave = t >> 5;
    const int wm   = wave & 3;      // row group: 32 rows each (4 groups = 128 rows)
    const int wn   = wave >> 2;     // col group: 64 cols each (2 groups = 128 cols)
    const int l15  = lane & 15;
    const int kh   = lane >> 4;     // half-wave selector per ISA A/B fragment layout

    if (t < BM) {
        s_sqi[t] = sq[bi * BM + t];
        s_li[t]  = labels[bi * BM + t];
    } else if (t < 2 * BM) {
        int u = t - BM;
        s_sqj[u] = sq[bj * BN + u];
        s_lj[u]  = labels[bj * BN + u];
    }

    v8f c00 = {0.f,0.f,0.f,0.f,0.f,0.f,0.f,0.f};
    v8f c01 = c00, c02 = c00, c03 = c00;
    v8f c10 = c00, c11 = c00, c12 = c00, c13 = c00;

    const int ldr = t >> 3;         // 0..31 (4 passes cover 128 rows)
    const int ldc = (t & 7) << 2;   // float4 column within BK

    const float* pa0 = &sA[(wm * 32 +      l15) * LSTR];
    const float* pa1 = &sA[(wm * 32 + 16 + l15) * LSTR];
    const float* pb0 = &sB[(wn * 64 +      l15) * LSTR];
    const float* pb1 = &sB[(wn * 64 + 16 + l15) * LSTR];
    const float* pb2 = &sB[(wn * 64 + 32 + l15) * LSTR];
    const float* pb3 = &sB[(wn * 64 + 48 + l15) * LSTR];

    for (int k0 = 0; k0 < D_LAT; k0 += BK) {
        __syncthreads();
#pragma unroll
        for (int p = 0; p < 4; ++p) {
            const int r = ldr + p * 32;
            const float4 va = *(const float4*)(X + (size_t)(bi * BM + r) * D_LAT + k0 + ldc);
            sA[r * LSTR + ldc + 0] = va.x;
            sA[r * LSTR + ldc + 1] = va.y;
            sA[r * LSTR + ldc + 2] = va.z;
            sA[r * LSTR + ldc + 3] = va.w;
            const float4 vb = *(const float4*)(X + (size_t)(bj * BN + r) * D_LAT + k0 + ldc);
            sB[r * LSTR + ldc + 0] = vb.x;
            sB[r * LSTR + ldc + 1] = vb.y;
            sB[r * LSTR + ldc + 2] = vb.z;
            sB[r * LSTR + ldc + 3] = vb.w;
        }
        __syncthreads();

#pragma unroll
        for (int kk = 0; kk < BK; kk += 4) {
            const int ko = kk + 2 * kh;   // VGPR0 -> K{0,1} lanes 0-15 / K{2,3} lanes 16-31
            v2f a0, a1, b0, b1, b2, b3;
            a0.x = pa0[ko]; a0.y = pa0[ko + 1];
            a1.x = pa1[ko]; a1.y = pa1[ko + 1];
            b0.x = pb0[ko]; b0.y = pb0[ko + 1];
            b1.x = pb1[ko]; b1.y = pb1[ko + 1];
            b2.x = pb2[ko]; b2.y = pb2[ko + 1];
            b3.x = pb3[ko]; b3.y = pb3[ko + 1];
            WSTEP(c00, a0, b0); WSTEP(c01, a0, b1); WSTEP(c02, a0, b2); WSTEP(c03, a0, b3);
            WSTEP(c10, a1, b0); WSTEP(c11, a1, b1); WSTEP(c12, a1, b2); WSTEP(c13, a1, b3);
        }
    }

    // Epilogue. C/D layout: VGPR v holds (M = v, lanes 0-15) and (M = v+8, lanes 16-31),
    // N = lane & 15 within each 16-col sub-tile.
    float pos = 0.f, pcnt = 0.f, neg = 0.f, ncnt = 0.f;

    auto accum_tile = [&](const v8f& c, int rl_base, int cl) {
        const int j = bj * BN + cl;
        const float sqj = s_sqj[cl];
        const int   lj  = s_lj[cl];
#pragma unroll
        for (int v = 0; v < 8; ++v) {
            const int rl = rl_base + v;
            const int i  = bi * BM + rl;
            if (i < j) {
                const float d2 = s_sqi[rl] + sqj - 2.0f * c[v];
                const float d  = (d2 > 0.f) ? sqrtf(fmaxf(d2, 1e-12f)) : 0.f;
                if (s_li[rl] == lj) { pos += fmaxf(d - POS_MARGIN, 0.f); pcnt += 1.f; }
                else                { neg += fmaxf(NEG_MARGIN - d, 0.f); ncnt += 1.f; }
            }
        }
    };

    const int rb0 = wm * 32 + kh * 8;        // row-group 0 fragment rows
    const int rb1 = wm * 32 + 16 + kh * 8;   // row-group 1 fragment rows
    const int cb  = wn * 64 + l15;
    accum_tile(c00, rb0, cb);      accum_tile(c01, rb0, cb + 16);
    accum_tile(c02, rb0, cb + 32); accum_tile(c03, rb0, cb + 48);
    accum_tile(c10, rb1, cb);      accum_tile(c11, rb1, cb + 16);
    accum_tile(c12, rb1, cb + 32); accum_tile(c13, rb1, cb + 48);

    red_x[t] = pos; red_y[t] = pcnt; red_z[t] = neg; red_w[t] = ncnt;
    __syncthreads();
    for (int off = 128; off > 0; off >>= 1) {
        if (t < off) {
            red_x[t] += red_x[t + off];
            red_y[t] += red_y[t + off];
            red_z[t] += red_z[t + off];
            red_w[t] += red_w[t + off];
        }
        __syncthreads();
    }
    if (t == 0) {
        atomicAdd(&acc[0], (double)red_x[0]);
        atomicAdd(&acc[1], (double)red_y[0]);
        atomicAdd(&acc[2], (double)red_z[0]);
        atomicAdd(&acc[3], (double)red_w[0]);
    }
}

__global__ __launch_bounds__(256) void mse_kernel(
    const float* __restrict__ r, const float* __restrict__ o, double* __restrict__ acc) {
    __shared__ float red[256];
    const size_t total = (size_t)N_ROWS * D_IN;
    float s = 0.f;
    for (size_t idx = (size_t)blockIdx.x * 256 + threadIdx.x; idx < total;
         idx += (size_t)gridDim.x * 256) {
        const float d = r[idx] - o[idx];
        s = fmaf(d, d, s);
    }
    red[threadIdx.x] = s;
    __syncthreads();
    for (int off = 128; off > 0; off >>= 1) {
        if (threadIdx.x < off) red[threadIdx.x] += red[threadIdx.x + off];
        __syncthreads();
    }
    if (threadIdx.x == 0) atomicAdd(&acc[4], (double)red[0]);
}

__global__ void finalize_kernel(const double* __restrict__ acc, float* __restrict__ out) {
    const double contrast = acc[0] / acc[1] + acc[2] / acc[3];
    const double mse = acc[4] / ((double)N_ROWS * (double)D_IN);
    out[0] = (float)(contrast + 0.3 * mse);
}

extern "C" void kernel_launch(void* const* d_in, const int* in_sizes, int n_in,
                              void* d_out, int out_size, void* d_ws, size_t ws_size,
                              hipStream_t stream) {
    const float* X      = (const float*)d_in[0];
    const int*   labels = (const int*)d_in[1];
    const float* rec    = (const float*)d_in[2];
    const float* orig   = (const float*)d_in[3];
    float* out = (float*)d_out;

    double* acc = (double*)d_ws;
    float*  sq  = (float*)((char*)d_ws + 64);

    zero_accum_kernel<<<1, 32, 0, stream>>>(acc);
    row_sqnorm_kernel<<<N_ROWS, 256, 0, stream>>>(X, sq);
    dim3 grid(N_ROWS / BN, N_ROWS / BM);
    gram_loss_kernel<<<grid, 256, 0, stream>>>(X, sq, labels, acc);
    mse_kernel<<<1024, 256, 0, stream>>>(rec, orig, acc);
    finalize_kernel<<<1, 1, 0, stream>>>(acc, out);
}